// CCLoss_10359461118288
// MI455X (gfx1250) — compile-verified
//
#include <hip/hip_runtime.h>
#include <hip/hip_bf16.h>

typedef __attribute__((ext_vector_type(2))) float v2f;
typedef __attribute__((ext_vector_type(8))) float v8f;
typedef unsigned long long u64;

#define BS 8
#define LN 64
#define IMG 256
#define NPIX (IMG*IMG)
#define TOPK 5
#define CHUNKS 32        // blocks per image
#define WAVES 8          // waves per block (256 threads, wave32)
#define TILES 16         // 16-pixel tiles per wave -> 8*16*16 = 2048 px/block, *32 = 65536

// ---------------------------------------------------------------------------
// Monotone float -> uint key (order-preserving for all finite floats).
// ---------------------------------------------------------------------------
__device__ __forceinline__ unsigned int fkey(float v) {
    unsigned int u = __float_as_uint(v);
    return u ^ ((unsigned int)((int)u >> 31) | 0x80000000u);
}

// ---------------------------------------------------------------------------
// Sorted top-5 insert on packed (value,index) u64 keys, ascending.
// Single u64 ordering == (value asc, index asc) tie-break, exactly
// jax.lax.top_k(-sim) selection order. Because K[] is sorted, insertion is
// a pure compare-exchange chain: 5 x (v_min_u64 + v_max_u64), no cmp/cndmask.
// ---------------------------------------------------------------------------
__device__ __forceinline__ void insert5(u64 k, u64 K[TOPK]) {
#pragma unroll
    for (int j = 0; j < TOPK; ++j) {
        u64 lo = (k < K[j]) ? k : K[j];   // v_min_u64
        u64 hi = (k < K[j]) ? K[j] : k;   // v_max_u64
        K[j] = lo;
        k = hi;
    }
}

// ---------------------------------------------------------------------------
// K1: grid_sample(nearest, align_corners=False, zeros) + build WMMA B-matrix.
// Output pair (b,l) takes its position from prediction flat row n = l*8 + b
// and samples image b (reference's  img_idx = n % bs  +  (L,bs) transpose).
// Bmat[b][k][l]: k<3 -> -2*pooled_c, k==3 -> 1.0   (so A*B = |img|^2 - 2<img,c>)
// ---------------------------------------------------------------------------
__global__ __launch_bounds__(256) void cc_sample_kernel(
        const float* __restrict__ preds, const float* __restrict__ img,
        float* __restrict__ Bmat) {
    int n = blockIdx.x * blockDim.x + threadIdx.x;
    if (n >= BS * LN) return;
    int b = n & (BS - 1);
    int l = n >> 3;
    float px = preds[n * 8 + 0];
    float py = preds[n * 8 + 1];
    float cx = px * (float)IMG - 0.5f;
    float cy = py * (float)IMG - 0.5f;
    int ix = (int)rintf(cx);               // v_rndne_f32: round-half-even (jnp.rint)
    int iy = (int)rintf(cy);
    bool valid = (ix >= 0) && (ix < IMG) && (iy >= 0) && (iy < IMG);
    int ixc = min(max(ix, 0), IMG - 1);
    int iyc = min(max(iy, 0), IMG - 1);
    float m = valid ? -2.0f : 0.0f;        // fold "-2" and zero-padding together
    size_t pofs = (size_t)iyc * IMG + ixc;
    float c0 = img[((size_t)(b * 3 + 0) << 16) + pofs];
    float c1 = img[((size_t)(b * 3 + 1) << 16) + pofs];
    float c2 = img[((size_t)(b * 3 + 2) << 16) + pofs];
    Bmat[(b * 4 + 0) * LN + l] = m * c0;
    Bmat[(b * 4 + 1) * LN + l] = m * c1;
    Bmat[(b * 4 + 2) * LN + l] = m * c2;
    Bmat[(b * 4 + 3) * LN + l] = 1.0f;
}

// ---------------------------------------------------------------------------
// K2: WMMA similarity + per-lane top-5 + block merge.
// One block = (image b, pixel chunk). Each wave: B fragments for all 64 l's
// (4 groups of 16) live in registers; stream 16-pixel A tiles through
// v_wmma_f32_16x16x4_f32. C layout: lane holds l = g*16 + (lane&15) and
// pixel rows M = r + (lane>=16 ? 8 : 0).
// ---------------------------------------------------------------------------
__global__ __launch_bounds__(256) void cc_sim_topk_kernel(
        const float* __restrict__ img, const float* __restrict__ Bmat,
        u64* __restrict__ cand) {
    __shared__ u64 sKey[LN][WAVES * 10];     // 80 candidates per l per block (40KB)

    const int b     = blockIdx.y;
    const int chunk = blockIdx.x;
    const int wave  = threadIdx.x >> 5;
    const int lane  = threadIdx.x & 31;
    const int hv    = lane >> 4;             // half-wave select (K01 vs K23)
    const int hl    = lane & 15;

    // --- B fragments: 4x16 f32, VGPR0 = {K0 | K2}, VGPR1 = {K1 | K3} ---
    const float* Bb = Bmat + b * 4 * LN;
    v2f Bf[4];
#pragma unroll
    for (int g = 0; g < 4; ++g) {
        int l = g * 16 + hl;
        Bf[g][0] = Bb[(hv ? 2 : 0) * LN + l];
        Bf[g][1] = Bb[(hv ? 3 : 1) * LN + l];
    }

    u64 Kk[4][TOPK];
#pragma unroll
    for (int g = 0; g < 4; ++g)
#pragma unroll
        for (int k = 0; k < TOPK; ++k) Kk[g][k] = ~0ULL;   // +inf sentinel

    const float* imgR = img + ((size_t)(b * 3 + 0) << 16);
    const float* imgG = img + ((size_t)(b * 3 + 1) << 16);
    const float* imgB = img + ((size_t)(b * 3 + 2) << 16);
    const int base = chunk * (WAVES * TILES * 16) + wave * (TILES * 16);

    for (int t = 0; t < TILES; ++t) {
        int pix = base + t * 16 + hl;
        if (t + 1 < TILES) {                 // gfx1250 global_prefetch_b8, near-cache
            __builtin_prefetch(imgR + pix + 16, 0, 3);
            __builtin_prefetch(imgG + pix + 16, 0, 3);
            __builtin_prefetch(imgB + pix + 16, 0, 3);
        }
        float r  = imgR[pix];
        float g  = imgG[pix];
        float bl = imgB[pix];
        float n2 = r * r + g * g + bl * bl;  // K3 row: |img_p|^2
        // A fragment 16x4 f32: VGPR0 = {K0=R | K2=B}, VGPR1 = {K1=G | K3=|.|^2}
        v2f a;
        a[0] = hv ? bl : r;
        a[1] = hv ? n2 : g;
#pragma unroll
        for (int grp = 0; grp < 4; ++grp) {
            v8f acc = {};
            acc = __builtin_amdgcn_wmma_f32_16x16x4_f32(
                false, a, false, Bf[grp], (short)0, acc, false, false);
#pragma unroll
            for (int rr = 0; rr < 8; ++rr) {
                int prow = rr + hv * 8;
                int pidx = base + t * 16 + prow;   // linear y*256+x, matches reshape
                u64 key = ((u64)fkey(acc[rr]) << 32) | (unsigned int)pidx;
                insert5(key, Kk[grp]);
            }
        }
    }

    // --- dump per-lane lists; lanes (i, i+16) own the same l, different rows ---
#pragma unroll
    for (int g = 0; g < 4; ++g) {
        int l = g * 16 + hl;
#pragma unroll
        for (int k = 0; k < TOPK; ++k)
            sKey[l][wave * 10 + hv * 5 + k] = Kk[g][k];
    }
    __syncthreads();

    // --- 80 -> 5 merge per l, one thread per l ---
    if (threadIdx.x < LN) {
        int l = threadIdx.x;
        u64 M[TOPK];
#pragma unroll
        for (int k = 0; k < TOPK; ++k) M[k] = ~0ULL;
        for (int s = 0; s < WAVES * 10; ++s)
            insert5(sKey[l][s], M);
        int off = ((b * LN + l) * CHUNKS + chunk) * TOPK;
#pragma unroll
        for (int k = 0; k < TOPK; ++k) cand[off + k] = M[k];
    }
}

// ---------------------------------------------------------------------------
// K3: global top-5 per (b,l) over CHUNKS*5 candidate keys -> target coords.
// tgt stored in the same (value-ascending) order as the reference's top_k.
// ---------------------------------------------------------------------------
__global__ __launch_bounds__(256) void cc_select_kernel(
        const u64* __restrict__ cand, float* __restrict__ tgt) {
    int pair = blockIdx.x * blockDim.x + threadIdx.x;
    if (pair >= BS * LN) return;
    u64 M[TOPK];
#pragma unroll
    for (int k = 0; k < TOPK; ++k) M[k] = ~0ULL;
    int off = pair * CHUNKS * TOPK;
    for (int s = 0; s < CHUNKS * TOPK; ++s)
        insert5(cand[off + s], M);
#pragma unroll
    for (int k = 0; k < TOPK; ++k) {
        int idx = (int)(unsigned int)(M[k] & 0xffffffffu);
        tgt[(pair * TOPK + k) * 2 + 0] = (float)(idx & (IMG - 1)) * (1.0f / IMG);
        tgt[(pair * TOPK + k) * 2 + 1] = (float)(idx >> 8)        * (1.0f / IMG);
    }
}

// ---------------------------------------------------------------------------
// K4: roll(1, axis=l) + argmin over K (first-min) + MSE over l>=1, mean /504.
// Single workgroup, LDS tree reduction, scalar out.
// ---------------------------------------------------------------------------
__global__ __launch_bounds__(512) void cc_loss_kernel(
        const float* __restrict__ preds, const float* __restrict__ tgt,
        float* __restrict__ out) {
    __shared__ float sdata[BS * LN];
    int t = threadIdx.x;                 // 0..511
    int b = t >> 6, l = t & (LN - 1);
    float contrib = 0.0f;
    if (l >= 1) {
        float px = preds[(b * LN + l) * 8 + 0];
        float py = preds[(b * LN + l) * 8 + 1];
        const float* tp = tgt + (b * LN + (l - 1)) * TOPK * 2;  // tgt_down[b,l]
        float best = 3.4e38f, bx = 0.0f, by = 0.0f;
#pragma unroll
        for (int k = 0; k < TOPK; ++k) {
            float tx = tp[k * 2 + 0], ty = tp[k * 2 + 1];
            float dx = px - tx, dy = py - ty;
            float d = dx * dx + dy * dy;
            if (d < best) { best = d; bx = tx; by = ty; }  // first-min tie-break
        }
        float ex = px - bx, ey = py - by;
        contrib = ex * ex + ey * ey;
    }
    sdata[t] = contrib;
    __syncthreads();
    for (int s = 256; s > 0; s >>= 1) {
        if (t < s) sdata[t] += sdata[t + s];
        __syncthreads();
    }
    if (t == 0) out[0] = sdata[0] * (1.0f / (BS * (LN - 1)));
}

// ---------------------------------------------------------------------------
extern "C" void kernel_launch(void* const* d_in, const int* in_sizes, int n_in,
                              void* d_out, int out_size, void* d_ws, size_t ws_size,
                              hipStream_t stream) {
    const float* preds = (const float*)d_in[0];   // (8,64,8) f32
    const float* img   = (const float*)d_in[1];   // (8,3,256,256) f32
    char* ws = (char*)d_ws;

    // workspace layout
    float* Bmat = (float*)ws;                     // 8*4*64 f32      =   8 KB
    u64*   cand = (u64*)(ws + 8192);              // 512*32*5 u64    = 640 KB
    float* tgt  = (float*)(ws + 8192 + 655360);   // 512*5*2 f32     =  20 KB
    float* outp = (float*)d_out;

    cc_sample_kernel  <<<dim3(2),          dim3(256), 0, stream>>>(preds, img, Bmat);
    cc_sim_topk_kernel<<<dim3(CHUNKS, BS), dim3(256), 0, stream>>>(img, Bmat, cand);
    cc_select_kernel  <<<dim3(2),          dim3(256), 0, stream>>>(cand, tgt);
    cc_loss_kernel    <<<dim3(1),          dim3(512), 0, stream>>>(preds, tgt, outp);
}